// _Self_Attention_Adj_867583394376
// MI455X (gfx1250) — compile-verified
//
#include <hip/hip_runtime.h>
#include <hip/hip_bf16.h>
#include <math.h>

// Problem constants (from reference setup_inputs)
#define Bq 4
#define Fq 256
#define Nq 4096
#define Aq 64
#define NEG_SLOPE 0.01f
#define NSPLIT 16                 // n-axis splits per column tile for occupancy
#define NCHUNK (Nq / NSPLIT)      // 256 rows per split
#define MT 2                      // m-tiles (16 cols each) per wave

typedef __attribute__((ext_vector_type(2))) float v2f;
typedef __attribute__((ext_vector_type(8))) float v8f;

// fp32 WMMA 16x16x4:  D(16x16 f32) = A(16x4 f32) * B(4x16 f32) + C
// per-lane: row/col = lane%16 ; K pair = {0,1} for lanes 0-15, {2,3} for 16-31
// C/D: VGPR j -> row j (lanes 0-15), row j+8 (lanes 16-31)
__device__ __forceinline__ v8f wmma_f32(v2f a, v2f b, v8f c) {
    return __builtin_amdgcn_wmma_f32_16x16x4_f32(false, a, false, b, (short)0, c,
                                                 false, false);
}

__device__ __forceinline__ float leaky(float v) {
    return v >= 0.0f ? v : NEG_SLOPE * v;
}

// online softmax update with the 8 row-values this lane holds for its column
__device__ __forceinline__ void online_update(const v8f& acc, float& run_m,
                                              float& run_s) {
    float tm = acc[0];
#pragma unroll
    for (int j = 1; j < 8; ++j) tm = fmaxf(tm, acc[j]);
    const float nm = fmaxf(run_m, tm);
    float s = run_s * __expf(run_m - nm);
#pragma unroll
    for (int j = 0; j < 8; ++j) s += __expf(acc[j] - nm);
    run_m = nm;
    run_s = s;
}

// fold lanes L and L+16 (same column, complementary row groups)
__device__ __forceinline__ void merge_halves(float& M, float& S, float run_m,
                                             float run_s) {
    const float om = __shfl_xor(run_m, 16, 32);
    const float os = __shfl_xor(run_s, 16, 32);
    M = fmaxf(run_m, om);
    S = run_s * __expf(run_m - M) + os * __expf(om - M);
}

// ---------------------------------------------------------------------------
// Kernel 1: Q[b,n,a] = leakyrelu(sum_f x[b,f,n] * W[f,a])  for W in {queue,key}
// One wave per 16x16 (n,a) output tile. 8 waves per block.
// ---------------------------------------------------------------------------
__global__ void qk_proj_kernel(const float* __restrict__ x,
                               const float* __restrict__ queue,
                               const float* __restrict__ keyw,
                               float* __restrict__ Qo,
                               float* __restrict__ Ko) {
    const int lane = threadIdx.x & 31;
    const int wid  = blockIdx.x * (blockDim.x >> 5) + (threadIdx.x >> 5);
    const int lr = lane & 15;          // row (A) / col (B,C)
    const int hi = lane >> 4;          // K-pair selector

    // task decode: [isK][b][ntile (256)][atile (4)]
    const int perMat = Bq * (Nq / 16) * (Aq / 16);   // 4096
    const int isK   = wid >= perMat;
    int t = wid - isK * perMat;
    const int b     = t >> 10;                 // /1024
    const int r     = t & 1023;
    const int n0    = (r >> 2) << 4;           // ntile*16
    const int a0    = (r & 3) << 4;            // atile*16

    const float* W   = isK ? keyw : queue;
    float*       out = isK ? Ko   : Qo;

    const float* xbase = x + (size_t)b * Fq * Nq + n0 + lr;   // + f*Nq
    const float* wbase = W + a0 + lr;                          // + f*Aq

    v8f acc = {};
#pragma unroll 4
    for (int k0 = 0; k0 < Fq; k0 += 4) {
        const int f = k0 + 2 * hi;
        v2f av, bv;
        av.x = xbase[(size_t)f * Nq];
        av.y = xbase[(size_t)(f + 1) * Nq];
        bv.x = wbase[f * Aq];
        bv.y = wbase[(f + 1) * Aq];
        acc = wmma_f32(av, bv, acc);
    }

    float* orow = out + ((size_t)b * Nq + n0 + 8 * hi) * Aq + a0 + lr;
#pragma unroll
    for (int j = 0; j < 8; ++j) {
        orow[(size_t)j * Aq] = leaky(acc[j]);
    }
}

// ---------------------------------------------------------------------------
// Kernel 2: streaming online-softmax stats per column m (softmax over n).
// One wave handles (b, 32-col tile pair, n-split): sweeps NCHUNK rows with
// two register-resident K^T tiles and two independent WMMA accumulators;
// each Q tile load feeds both accumulators (halves L2 read traffic).
// ---------------------------------------------------------------------------
__global__ void scores_stats_kernel(const float* __restrict__ Q,
                                    const float* __restrict__ K,
                                    float* __restrict__ pmax,
                                    float* __restrict__ psum) {
    const int lane = threadIdx.x & 31;
    const int wid  = blockIdx.x * (blockDim.x >> 5) + (threadIdx.x >> 5);
    const int lr = lane & 15;
    const int hi = lane >> 4;

    // task decode: [b][mpair (128)][split (16)]
    const int b     = wid >> 11;               // / (128*16)
    const int rem   = wid & 2047;
    const int m0    = (rem >> 4) << 5;         // 32-column pair base
    const int split = rem & 15;
    const int nlo   = split * NCHUNK;

    // B operands: two K^T tiles, resident in registers for the whole sweep
    const float* krow = K + ((size_t)b * Nq + m0 + lr) * Aq + 2 * hi;
    v2f kb0[16], kb1[16];
#pragma unroll
    for (int c = 0; c < 16; ++c) {
        kb0[c] = *(const v2f*)(krow + 4 * c);
        kb1[c] = *(const v2f*)(krow + 16 * Aq + 4 * c);
    }

    float run_m0 = -3.0e38f, run_s0 = 0.0f;
    float run_m1 = -3.0e38f, run_s1 = 0.0f;

    for (int n0 = nlo; n0 < nlo + NCHUNK; n0 += 16) {
        const float* qrow = Q + ((size_t)b * Nq + n0 + lr) * Aq + 2 * hi;
        v8f acc0 = {}, acc1 = {};
#pragma unroll
        for (int c = 0; c < 16; ++c) {
            v2f qa = *(const v2f*)(qrow + 4 * c);
            acc0 = wmma_f32(qa, kb0[c], acc0);
            acc1 = wmma_f32(qa, kb1[c], acc1);
        }
        online_update(acc0, run_m0, run_s0);
        online_update(acc1, run_m1, run_s1);
    }

    float M0, S0, M1, S1;
    merge_halves(M0, S0, run_m0, run_s0);
    merge_halves(M1, S1, run_m1, run_s1);

    if (hi == 0) {
        const size_t i0 = ((size_t)b * Nq + m0 + lr) * NSPLIT + split;
        const size_t i1 = ((size_t)b * Nq + m0 + 16 + lr) * NSPLIT + split;
        pmax[i0] = M0;
        psum[i0] = S0;
        pmax[i1] = M1;
        psum[i1] = S1;
    }
}

// ---------------------------------------------------------------------------
// Kernel 2b: fold the NSPLIT partial (max, sum) pairs per column.
// ---------------------------------------------------------------------------
__global__ void stats_combine_kernel(const float* __restrict__ pmax,
                                     const float* __restrict__ psum,
                                     float* __restrict__ colmax,
                                     float* __restrict__ colinv) {
    const int tid = blockIdx.x * blockDim.x + threadIdx.x;
    if (tid >= Bq * Nq) return;
    float M = -3.0e38f;
#pragma unroll
    for (int s = 0; s < NSPLIT; ++s) M = fmaxf(M, pmax[tid * NSPLIT + s]);
    float S = 0.0f;
#pragma unroll
    for (int s = 0; s < NSPLIT; ++s)
        S += psum[tid * NSPLIT + s] * __expf(pmax[tid * NSPLIT + s] - M);
    colmax[tid] = M;
    colinv[tid] = 1.0f / S;
}

// ---------------------------------------------------------------------------
// Kernel 3: recompute score tiles, normalize, single streaming (NT) write.
// out[b,n,m] = exp(S[b,n,m] - colmax[b,m]) * colinv[b,m]
// ---------------------------------------------------------------------------
__global__ void scores_write_kernel(const float* __restrict__ Q,
                                    const float* __restrict__ K,
                                    const float* __restrict__ colmax,
                                    const float* __restrict__ colinv,
                                    float* __restrict__ out) {
    const int lane = threadIdx.x & 31;
    const int wid  = blockIdx.x * (blockDim.x >> 5) + (threadIdx.x >> 5);
    const int lr = lane & 15;
    const int hi = lane >> 4;

    const int b     = wid >> 11;
    const int rem   = wid & 2047;
    const int m0    = (rem >> 4) << 5;
    const int split = rem & 15;
    const int nlo   = split * NCHUNK;

    const float* krow = K + ((size_t)b * Nq + m0 + lr) * Aq + 2 * hi;
    v2f kb0[16], kb1[16];
#pragma unroll
    for (int c = 0; c < 16; ++c) {
        kb0[c] = *(const v2f*)(krow + 4 * c);
        kb1[c] = *(const v2f*)(krow + 16 * Aq + 4 * c);
    }

    const float cm0 = colmax[(size_t)b * Nq + m0 + lr];
    const float ci0 = colinv[(size_t)b * Nq + m0 + lr];
    const float cm1 = colmax[(size_t)b * Nq + m0 + 16 + lr];
    const float ci1 = colinv[(size_t)b * Nq + m0 + 16 + lr];

    for (int n0 = nlo; n0 < nlo + NCHUNK; n0 += 16) {
        const float* qrow = Q + ((size_t)b * Nq + n0 + lr) * Aq + 2 * hi;
        v8f acc0 = {}, acc1 = {};
#pragma unroll
        for (int c = 0; c < 16; ++c) {
            v2f qa = *(const v2f*)(qrow + 4 * c);
            acc0 = wmma_f32(qa, kb0[c], acc0);
            acc1 = wmma_f32(qa, kb1[c], acc1);
        }
        float* orow = out + ((size_t)b * Nq + n0 + 8 * hi) * Nq + m0 + lr;
#pragma unroll
        for (int j = 0; j < 8; ++j) {
            __builtin_nontemporal_store(__expf(acc0[j] - cm0) * ci0,
                                        orow + (size_t)j * Nq);
            __builtin_nontemporal_store(__expf(acc1[j] - cm1) * ci1,
                                        orow + (size_t)j * Nq + 16);
        }
    }
}

// ---------------------------------------------------------------------------
extern "C" void kernel_launch(void* const* d_in, const int* in_sizes, int n_in,
                              void* d_out, int out_size, void* d_ws, size_t ws_size,
                              hipStream_t stream) {
    (void)in_sizes; (void)n_in; (void)out_size; (void)ws_size;
    const float* x     = (const float*)d_in[0];   // [B,F,N]
    const float* queue = (const float*)d_in[1];   // [F,A]
    const float* keyw  = (const float*)d_in[2];   // [F,A]
    float* out = (float*)d_out;                   // [B,N,N]

    // workspace carve-up (floats): ~10.5 MB total
    float* ws = (float*)d_ws;
    float* Qm     = ws;                                   // B*N*A = 1,048,576
    float* Km     = Qm + (size_t)Bq * Nq * Aq;            // B*N*A
    float* pmax   = Km + (size_t)Bq * Nq * Aq;            // B*N*NSPLIT
    float* psum   = pmax + (size_t)Bq * Nq * NSPLIT;      // B*N*NSPLIT
    float* colmax = psum + (size_t)Bq * Nq * NSPLIT;      // B*N
    float* colinv = colmax + (size_t)Bq * Nq;             // B*N

    // K1: 2 * B * (N/16) * (A/16) = 8192 waves -> 1024 blocks x 256 thr
    qk_proj_kernel<<<1024, 256, 0, stream>>>(x, queue, keyw, Qm, Km);

    // K2: B * (N/32 m-pairs) * NSPLIT = 8192 waves -> 1024 blocks
    scores_stats_kernel<<<1024, 256, 0, stream>>>(Qm, Km, pmax, psum);

    // K2b: B*N = 16384 columns
    stats_combine_kernel<<<64, 256, 0, stream>>>(pmax, psum, colmax, colinv);

    // K3: same decomposition as K2, writes 256 MB output exactly once (NT)
    scores_write_kernel<<<1024, 256, 0, stream>>>(Qm, Km, colmax, colinv, out);
}